// ColorNodeModel_2843268350529
// MI455X (gfx1250) — compile-verified
//
#include <hip/hip_runtime.h>

#define D 128
#define WPB 4      // waves per block in the MLP kernel
#define HPAD 132   // LDS row stride (floats): 132 % 64 banks = 4 -> conflict-free A-frag reads

typedef float v2f __attribute__((ext_vector_type(2)));
typedef float v4f __attribute__((ext_vector_type(4)));
typedef float v8f __attribute__((ext_vector_type(8)));

// ---------------- kernel 1: zero the aggregation workspace ----------------
// regular (RT) stores: the zeros are immediately RMW'd by L2 atomics
__global__ __launch_bounds__(256) void zero_ws(v4f* __restrict__ p, int n4) {
    int i = blockIdx.x * blockDim.x + threadIdx.x;
    if (i < n4) p[i] = (v4f){0.f, 0.f, 0.f, 0.f};
}

// ---------------- kernel 2: scatter-add edge_attr to both endpoints -------
// one wave (32 lanes) per edge; lane covers 4 contiguous floats of the row.
// edge stream is 307MB read-once -> NT loads so it doesn't evict the 25.6MB
// aggregation buffer (which must stay resident in the 192MB L2 for the atomics).
__global__ __launch_bounds__(256) void scatter_edges(const float* __restrict__ ea,
                                                     const int* __restrict__ ei,
                                                     float* __restrict__ aggr,
                                                     int E_) {
    int e    = (blockIdx.x * blockDim.x + threadIdx.x) >> 5;
    int lane = threadIdx.x & 31;
    if (e >= E_) return;
    int r = __builtin_nontemporal_load(ei + e);        // edge_index[0][e]
    int c = __builtin_nontemporal_load(ei + E_ + e);   // edge_index[1][e]
    const v4f* src = (const v4f*)(ea + (size_t)e * D);
    v4f v = __builtin_nontemporal_load(src + lane);
    int base = lane * 4;
    float* pr = aggr + (size_t)r * D + base;
    float* pc = aggr + (size_t)c * D + base;
    atomicAdd(pr + 0, v.x); atomicAdd(pr + 1, v.y);
    atomicAdd(pr + 2, v.z); atomicAdd(pr + 3, v.w);
    atomicAdd(pc + 0, v.x); atomicAdd(pc + 1, v.y);
    atomicAdd(pc + 2, v.z); atomicAdd(pc + 3, v.w);
}

// ---------------- kernel 3: fused 2-layer MLP via f32 WMMA ----------------
// one wave computes a 16-row x 128-col output tile.
// GEMM1: A = [x | aggr] (16x256), B = W1 (256x128)  -> h (relu, +b1) via LDS
// GEMM2: A = h (16x128),          B = W2 (128x128)  -> out (+b2)
// x/aggr rows are read exactly once -> NT loads; W1/W2/b1/b2 are hot -> RT.
__global__ __launch_bounds__(WPB * 32) void mlp_fused(const float* __restrict__ x,
                                                      const float* __restrict__ aggr,
                                                      const float* __restrict__ W1,
                                                      const float* __restrict__ b1,
                                                      const float* __restrict__ W2,
                                                      const float* __restrict__ b2,
                                                      float* __restrict__ out,
                                                      int N_) {
    __shared__ float hlds[WPB][16][HPAD];

    const int lane  = threadIdx.x & 31;
    const int wave  = threadIdx.x >> 5;
    const int tile  = blockIdx.x * WPB + wave;
    const int nTile = (N_ + 15) >> 4;
    if (tile >= nTile) return;             // wave-uniform: EXEC stays all-ones

    const int m     = lane & 15;           // row (A) / col (B,C,D) within tile
    const int khalf = (lane >> 4) * 2;     // lanes 16-31 hold K+2,K+3
    const int hi8   = (lane >> 4) * 8;     // C/D: lanes 16-31 hold rows M+8
    const int row   = tile * 16 + m;
    const int rowc  = row < N_ ? row : N_ - 1;

    // per-column-tile biases (col = n*16 + m depends only on lane)
    float bias1[8], bias2[8];
#pragma unroll
    for (int n = 0; n < 8; ++n) {
        bias1[n] = b1[n * 16 + m];
        bias2[n] = b2[n * 16 + m];
    }

    const float* xrow = x    + (size_t)rowc * D;
    const float* arow = aggr + (size_t)rowc * D;

    v8f acc[8] = {};

    // ---- GEMM1: K = 256 (cols 0..127 from x, 128..255 from aggr) ----
    for (int k = 0; k < 2 * D; k += 4) {
        const int kb = k + khalf;          // even; never straddles the 128 seam
        v2f a;
        if (kb < D) {
            a = __builtin_nontemporal_load((const v2f*)(xrow + kb));
        } else {
            a = __builtin_nontemporal_load((const v2f*)(arow + (kb - D)));
        }
#pragma unroll
        for (int n = 0; n < 8; ++n) {
            v2f b;
            b.x = W1[(size_t)kb * D + n * 16 + m];
            b.y = W1[(size_t)(kb + 1) * D + n * 16 + m];
            acc[n] = __builtin_amdgcn_wmma_f32_16x16x4_f32(
                false, a, false, b, (short)0, acc[n], false, false);
        }
    }

    // ---- bias + relu, stage h into this wave's LDS region ----
#pragma unroll
    for (int n = 0; n < 8; ++n) {
#pragma unroll
        for (int r = 0; r < 8; ++r) {
            float h = acc[n][r] + bias1[n];
            hlds[wave][r + hi8][n * 16 + m] = h > 0.f ? h : 0.f;
        }
        acc[n] = (v8f){};                  // reuse accumulators for GEMM2
    }

    // ---- GEMM2: K = 128, A-frags re-read from LDS (conflict-free pad) ----
    for (int k = 0; k < D; k += 4) {
        const int kb = k + khalf;
        v2f a;
        a.x = hlds[wave][m][kb];
        a.y = hlds[wave][m][kb + 1];
#pragma unroll
        for (int n = 0; n < 8; ++n) {
            v2f b;
            b.x = W2[(size_t)kb * D + n * 16 + m];
            b.y = W2[(size_t)(kb + 1) * D + n * 16 + m];
            acc[n] = __builtin_amdgcn_wmma_f32_16x16x4_f32(
                false, a, false, b, (short)0, acc[n], false, false);
        }
    }

    // ---- +b2, NT store (out is written once, never re-read) ----
#pragma unroll
    for (int n = 0; n < 8; ++n) {
#pragma unroll
        for (int r = 0; r < 8; ++r) {
            const int orow = tile * 16 + r + hi8;
            if (orow < N_)
                __builtin_nontemporal_store(acc[n][r] + bias2[n],
                                            out + (size_t)orow * D + n * 16 + m);
        }
    }
}

// ---------------------------------------------------------------------------
extern "C" void kernel_launch(void* const* d_in, const int* in_sizes, int n_in,
                              void* d_out, int out_size, void* d_ws, size_t ws_size,
                              hipStream_t stream) {
    const float* x  = (const float*)d_in[0];
    const int*   ei = (const int*)  d_in[1];
    const float* ea = (const float*)d_in[2];
    const float* W1 = (const float*)d_in[3];
    const float* b1 = (const float*)d_in[4];
    const float* W2 = (const float*)d_in[5];
    const float* b2 = (const float*)d_in[6];
    float*       out  = (float*)d_out;
    float*       aggr = (float*)d_ws;   // N*D floats of scratch

    const int N_ = in_sizes[0] / D;
    const int E_ = in_sizes[2] / D;

    const int n4 = (N_ * D) / 4;
    zero_ws<<<(n4 + 255) / 256, 256, 0, stream>>>((v4f*)aggr, n4);

    // 8 edges per 256-thread block (one wave per edge)
    scatter_edges<<<(E_ + 7) / 8, 256, 0, stream>>>(ea, ei, aggr, E_);

    // 64 rows per block (4 waves x 16-row tiles)
    mlp_fused<<<(N_ + WPB * 16 - 1) / (WPB * 16), WPB * 32, 0, stream>>>(
        x, aggr, W1, b1, W2, b2, out, N_);
}